// PillarScatter_62096637165639
// MI455X (gfx1250) — compile-verified
//
#include <hip/hip_runtime.h>
#include <stdint.h>

#define GX 512
#define GY 512
#define NB 8
#define NV 16384
#define NC 64
#define PLANE (GX * GY)          // 262144 cells per (b, plane)
#define CSTRIDE 68               // padded floats per LDS cell row (272B: 16B-aligned, 2-way banks max)
#define WAVES_PER_BLOCK 4

// ---------------------------------------------------------------------------
// Kernel 1: initialize the voxel-id map to -1.
// The map lives in the channel-0 plane of d_out (out[b][0][y][x]); kernel 3
// reads each map entry before overwriting it, and only the owning wave ever
// touches a given (b,y,x) column, so there is no cross-wave race.
// ---------------------------------------------------------------------------
__global__ __launch_bounds__(256)
void k_init_map(int* __restrict__ out_i) {
  unsigned tid = blockIdx.x * 256u + threadIdx.x;   // over NB*PLANE/4 = 2^19
  unsigned b = tid >> 16;                           // PLANE/4 == 2^16 int4 per plane
  unsigned q = tid & 0xFFFFu;
  int4* p = (int4*)(out_i + (size_t)b * NC * PLANE);
  p[q] = make_int4(-1, -1, -1, -1);
}

// ---------------------------------------------------------------------------
// Kernel 2: scatter voxel indices into the map. atomicMax gives a
// deterministic winner for duplicate (y,x) (JAX scatter with duplicates is
// unspecified, so any deterministic winner is acceptable). Map is only 8MB,
// stays L2-resident for kernel 3.
// ---------------------------------------------------------------------------
__global__ __launch_bounds__(256)
void k_scatter_ids(const int* __restrict__ coords, int* out_i) {
  unsigned tid = blockIdx.x * 256u + threadIdx.x;   // over NB*NV = 2^17
  unsigned b = tid >> 14;                           // NV == 2^14
  int v = (int)(tid & (NV - 1u));
  const int* c = coords + (size_t)tid * 3;
  int x = c[0];
  int y = c[1];
  if (x >= 0 && x < GX && y >= 0 && y < GY) {
    int* plane = out_i + (size_t)b * NC * PLANE;
    atomicMax(&plane[y * GX + x], v);
  }
}

// ---------------------------------------------------------------------------
// Kernel 3: gather + channel transpose. One wave per 32 consecutive x cells.
// Valid lanes async-DMA their 256B feature row into LDS (CDNA5
// global_load_async_to_lds_b128, ASYNCcnt-tracked, EXEC-masked), wave waits
// with s_wait_asynccnt, then streams 64 channel rows as 128B-coalesced
// NON-TEMPORAL stores so the 512MB write-once output bypasses L2.
// ---------------------------------------------------------------------------
__global__ __launch_bounds__(32 * WAVES_PER_BLOCK)
void k_gather(const float* __restrict__ pf, float* out) {
  __shared__ __align__(16) float smem[WAVES_PER_BLOCK * 32 * CSTRIDE]; // 34816 B
  const unsigned lane = threadIdx.x & 31u;
  const unsigned wid  = threadIdx.x >> 5;
  const unsigned w    = blockIdx.x * WAVES_PER_BLOCK + wid;  // global wave id
  const unsigned xt   = w & 15u;                             // 16 x-tiles per row
  const unsigned y    = (w >> 4) & (GY - 1u);
  const unsigned b    = w >> 13;                             // 512*16 = 2^13 waves/batch
  const unsigned x    = (xt << 5) + lane;

  // Read this tile's voxel ids from the plane-0 map (before overwriting it).
  const int* map_plane = (const int*)(out + (size_t)b * NC * PLANE);
  const int id = map_plane[y * GX + x];
  const bool valid = id >= 0;

  float* rowp = &smem[(wid * 32u + lane) * CSTRIDE];
  // Low 32 bits of a generic LDS pointer are the LDS offset (aperture rule).
  const uint32_t ldsrow = (uint32_t)(uintptr_t)rowp;

  if (valid) {
    // 256B feature row of this lane's voxel, 16 x B128 async chunks,
    // memory -> LDS with no VGPR staging.
    uint64_t g = (uint64_t)(uintptr_t)(pf + ((size_t)b * NV + (size_t)id) * NC);
#pragma unroll
    for (int j = 0; j < 16; ++j) {
      asm volatile("global_load_async_to_lds_b128 %0, %1, off"
                   :: "v"(ldsrow + (uint32_t)(j * 16)), "v"(g + (uint64_t)(j * 16))
                   : "memory");
    }
  }
#if __has_builtin(__builtin_amdgcn_s_wait_asynccnt)
  __builtin_amdgcn_s_wait_asynccnt(0);
#else
  asm volatile("s_wait_asynccnt 0" ::: "memory");
#endif

  // 64 channel rows; each store is 32 lanes * 4B = 128B coalesced, streamed
  // past L2 (write-once data) via non-temporal hint.
  float* op = out + (size_t)b * NC * PLANE + (size_t)y * GX + x;
#pragma unroll 8
  for (int c = 0; c < NC; ++c) {
    float v = valid ? rowp[c] : 0.0f;
    __builtin_nontemporal_store(v, &op[(size_t)c * PLANE]);
  }
}

// ---------------------------------------------------------------------------
extern "C" void kernel_launch(void* const* d_in, const int* in_sizes, int n_in,
                              void* d_out, int out_size, void* d_ws, size_t ws_size,
                              hipStream_t stream) {
  (void)in_sizes; (void)n_in; (void)out_size; (void)d_ws; (void)ws_size;
  const float* pf     = (const float*)d_in[0];  // [B, V, C] f32
  const int*   coords = (const int*)d_in[1];    // [B, V, 3] i32
  float*       out    = (float*)d_out;          // [B, C, Y, X] f32

  // 1) voxel-id map (plane-0 of out) <- -1   (RT stores: keep map in L2)
  k_init_map<<<(NB * PLANE / 4) / 256, 256, 0, stream>>>((int*)out);
  // 2) scatter voxel ids (deterministic atomicMax winner)
  k_scatter_ids<<<(NB * NV) / 256, 256, 0, stream>>>(coords, (int*)out);
  // 3) gather + transpose: one fully-coalesced, NT-streamed 512MB output pass
  const int totalWaves = NB * GY * (GX / 32);   // 65536
  k_gather<<<totalWaves / WAVES_PER_BLOCK, 32 * WAVES_PER_BLOCK, 0, stream>>>(pf, out);
}